// CASSI_Layer_v2_58884001628467
// MI455X (gfx1250) — compile-verified
//
#include <hip/hip_runtime.h>
#include <cstdint>

#define MDIM    512
#define NDIM    512
#define LDIM    12
#define NTILE   32
#define BATCH   16
#define ROWF    (NDIM * LDIM)        // 6144 floats per image row
#define YLEN    (NDIM + LDIM - 1)    // 523
#define THREADS 256

typedef float vf4 __attribute__((ext_vector_type(4)));

__global__ __launch_bounds__(THREADS)
void cassi_row_kernel(const float* __restrict__ in,
                      const float* __restrict__ wr, const float* __restrict__ wg,
                      const float* __restrict__ wb, const float* __restrict__ wc,
                      float* __restrict__ Xout, float* __restrict__ Yout)
{
    __shared__ __align__(16) float s_in[ROWF];       // 24576 B: one input row
    __shared__ float s_mask[NTILE * LDIM];           // 1536 B: mask tile row (32 x 12)
    __shared__ float s_Y[YLEN];                      // 2092 B: sheared integral

    const int tid = threadIdx.x;
    const int row = blockIdx.x;          // b*512 + m
    const int m32 = row & 31;            // m % 32 (512 % 32 == 0)

    // ---- Stage A: async DMA global -> LDS of the whole input row ------------
    // 6144 floats = 1536 x b128; 6 transfers per thread. CDNA5 async path:
    // data goes straight to LDS (ASYNCcnt-tracked), no VGPR staging. Input is
    // single-use streaming (201 MB > 192 MB L2) -> non-temporal hint.
    {
        const float* grow = in + (size_t)row * ROWF;
        unsigned lds_base = (unsigned)(uintptr_t)&s_in[0];  // low 32b = LDS byte offset
        #pragma unroll
        for (int c = 0; c < 6; ++c) {
            unsigned voff = (unsigned)(tid + c * THREADS) * 16u;
            unsigned ldsa = lds_base + voff;
            asm volatile("global_load_async_to_lds_b128 %0, %1, %2 th:TH_LOAD_NT"
                         :: "v"(ldsa), "v"(voff), "s"(grow) : "memory");
        }
    }

    // ---- Stage B (overlaps the async DMA): spectral mask tile ---------------
    // aux = (wr*fr + wg*fg + wb*fb + wc*fc) / (wr+wg+wb+wc), f* = Gaussians
    // over wl = linspace(420, 660, 12) nm.
    if (tid < NTILE * LDIM) {
        const int n32 = tid / LDIM;
        const int i   = tid - n32 * LDIM;
        const float wl = 420.0f + 240.0f * (float)i / 11.0f;
        const float dr = (wl - 620.0f) / 40.0f;
        const float dg = (wl - 530.0f) / 40.0f;
        const float db = (wl - 450.0f) / 40.0f;
        const float dc = (wl - 500.0f) / 60.0f;
        const float fr = expf(-0.5f * dr * dr);
        const float fg = expf(-0.5f * dg * dg);
        const float fb = expf(-0.5f * db * db);
        const float fc = expf(-0.5f * dc * dc);
        const int wi = m32 * NTILE + n32;
        const float r = wr[wi], g = wg[wi], b = wb[wi], c = wc[wi];
        s_mask[tid] = (r * fr + g * fg + b * fb + c * fc) / (r + g + b + c);
    }

    asm volatile("s_wait_asynccnt 0x0" ::: "memory");   // this wave's DMA done
    __syncthreads();                                    // all waves' DMA visible

    // ---- Stage C: shear + spectral integration ------------------------------
    // Y[n] = sum_i mask[(n-i)%32][i] * in[n-i][i], zero outside [0, NDIM)
    for (int n = tid; n < YLEN; n += THREADS) {
        float acc = 0.0f;
        #pragma unroll
        for (int i = 0; i < LDIM; ++i) {
            const int src = n - i;
            if (src >= 0 && src < NDIM)
                acc += s_mask[(src & 31) * LDIM + i] * s_in[src * LDIM + i];
        }
        s_Y[n] = acc;
    }
    __syncthreads();

    // ---- Stage D: adjoint un-shear + remask, streamed out -------------------
    // X[n][i] = mask[n%32][i] * Y[n+i]; flat layout so consecutive lanes store
    // consecutive 16B chunks (each wave b128 store = 512 B = 8 full lines).
    // One div per 4-element chunk; (n,i) carried incrementally.
    float* xrow = Xout + (size_t)row * ROWF;
    #pragma unroll
    for (int c = 0; c < 6; ++c) {
        const int f = (tid + c * THREADS) * 4;   // flat index n*12+i, 16B aligned
        int n = f / LDIM;
        int i = f - n * LDIM;
        vf4 v;
        #pragma unroll
        for (int k = 0; k < 4; ++k) {
            v[k] = s_mask[(n & 31) * LDIM + i] * s_Y[n + i];
            if (++i == LDIM) { i = 0; ++n; }
        }
        __builtin_nontemporal_store(v, (vf4*)(xrow + f));
    }

    float* yrow = Yout + (size_t)row * YLEN;
    for (int n = tid; n < YLEN; n += THREADS)
        __builtin_nontemporal_store(s_Y[n], yrow + n);
}

extern "C" void kernel_launch(void* const* d_in, const int* in_sizes, int n_in,
                              void* d_out, int out_size, void* d_ws, size_t ws_size,
                              hipStream_t stream) {
    const float* in = (const float*)d_in[0];
    const float* wr = (const float*)d_in[1];
    const float* wg = (const float*)d_in[2];
    const float* wb = (const float*)d_in[3];
    const float* wc = (const float*)d_in[4];

    float* X = (float*)d_out;                                   // (16,512,512,12)
    float* Y = X + (size_t)BATCH * MDIM * NDIM * LDIM;          // (16,512,523,1)

    dim3 grid(BATCH * MDIM);   // 8192 rows
    dim3 block(THREADS);
    cassi_row_kernel<<<grid, block, 0, stream>>>(in, wr, wg, wb, wc, X, Y);
}